// GaussianRasterizer_58334245814745
// MI455X (gfx1250) — compile-verified
//
#include <hip/hip_runtime.h>
#include <hip/hip_bf16.h>
#include <math.h>

// ---------------------------------------------------------------------------
// Gaussian splat forward renderer for gfx1250 (MI455X).
// Pipeline: preprocess -> bitonic sort+gather -> WMMA-tiled blend.
// power(pixel,gaussian) is a K=6 dot product -> V_WMMA_F32_16X16X4_F32 (x2, K padded to 8).
// ---------------------------------------------------------------------------

typedef float v2f __attribute__((ext_vector_type(2)));
typedef float v8f __attribute__((ext_vector_type(8)));

#define PN    2048
#define WIMG  112
#define HIMG  112
#define NPIX  (WIMG * HIMG)

// workspace layout (floats)
//   featU : [0        , PN*8 )   unsorted gaussian power-features (6 + 2 pad)
//   blendU: [PN*8     , PN*16)   unsorted {opa, r, g, b, depth, visflag, 0, 0}
//   featS : [PN*16    , PN*24)   sorted features
//   blendS: [PN*24    , PN*32)   sorted blend data
//   dkey  : [PN*32    , PN*33)   sort keys (depth or +inf)

__constant__ const float C0f = 0.28209479177387814f;
__constant__ const float C1f = 0.4886025119029199f;

// ---------------------------------------------------------------------------
__global__ void __launch_bounds__(256) preprocess_kernel(
    const float* __restrict__ means3D, const float* __restrict__ shs,
    const float* __restrict__ opac,    const float* __restrict__ scales,
    const float* __restrict__ smod_p,  const float* __restrict__ rots,
    const float* __restrict__ vm,      const float* __restrict__ pm,
    const float* __restrict__ campos,  const float* __restrict__ tfx_p,
    const float* __restrict__ tfy_p,   const int* __restrict__ deg_p,
    const int* __restrict__ w_p,       const int* __restrict__ h_p,
    float* __restrict__ ws,            float* __restrict__ out)
{
    int i = blockIdx.x * blockDim.x + threadIdx.x;
    if (i >= PN) return;

    float mx = means3D[i*3+0], my = means3D[i*3+1], mz = means3D[i*3+2];
    float tanfx = tfx_p[0], tanfy = tfy_p[0];
    float width = (float)w_p[0], height = (float)h_p[0];

    // p_view = [m,1] @ viewmatrix (row-vector convention)
    float pv0 = mx*vm[0] + my*vm[4] + mz*vm[8]  + vm[12];
    float pv1 = mx*vm[1] + my*vm[5] + mz*vm[9]  + vm[13];
    float pv2 = mx*vm[2] + my*vm[6] + mz*vm[10] + vm[14];
    float depth = pv2;

    // p_hom = [m,1] @ projmatrix ; p_proj = p_hom[:3] / (w + 1e-7)
    float ph0 = mx*pm[0] + my*pm[4] + mz*pm[8]  + pm[12];
    float ph1 = mx*pm[1] + my*pm[5] + mz*pm[9]  + pm[13];
    float ph3 = mx*pm[3] + my*pm[7] + mz*pm[11] + pm[15];
    float invw = 1.0f / (ph3 + 1e-7f);
    float pp0 = ph0 * invw, pp1 = ph1 * invw;

    float focal_y = height / (2.0f * tanfy);
    float focal_x = width  / (2.0f * tanfx);

    // quat -> rotation (normalized)
    float qw = rots[i*4+0], qx = rots[i*4+1], qy = rots[i*4+2], qz = rots[i*4+3];
    float qn = sqrtf(qw*qw + qx*qx + qy*qy + qz*qz);
    qw /= qn; qx /= qn; qy /= qn; qz /= qn;
    float R00 = 1.0f - 2.0f*(qy*qy + qz*qz), R01 = 2.0f*(qx*qy - qw*qz), R02 = 2.0f*(qx*qz + qw*qy);
    float R10 = 2.0f*(qx*qy + qw*qz), R11 = 1.0f - 2.0f*(qx*qx + qz*qz), R12 = 2.0f*(qy*qz - qw*qx);
    float R20 = 2.0f*(qx*qz - qw*qy), R21 = 2.0f*(qy*qz + qw*qx), R22 = 1.0f - 2.0f*(qx*qx + qy*qy);

    float sm = smod_p[0];
    float s0 = scales[i*3+0]*sm, s1 = scales[i*3+1]*sm, s2 = scales[i*3+2]*sm;
    float M00 = R00*s0, M01 = R01*s1, M02 = R02*s2;
    float M10 = R10*s0, M11 = R11*s1, M12 = R12*s2;
    float M20 = R20*s0, M21 = R21*s1, M22 = R22*s2;
    float Sg00 = M00*M00 + M01*M01 + M02*M02;
    float Sg01 = M00*M10 + M01*M11 + M02*M12;
    float Sg02 = M00*M20 + M01*M21 + M02*M22;
    float Sg11 = M10*M10 + M11*M11 + M12*M12;
    float Sg12 = M10*M20 + M11*M21 + M12*M22;
    float Sg22 = M20*M20 + M21*M21 + M22*M22;

    // Jacobian with FOV clipping
    float tz = pv2;
    tz = (fabsf(tz) < 1e-6f) ? 1e-6f : tz;
    float limx = 1.3f * tanfx, limy = 1.3f * tanfy;
    float txc = fminf(fmaxf(pv0 / tz, -limx), limx) * tz;
    float tyc = fminf(fmaxf(pv1 / tz, -limy), limy) * tz;
    float J00 = focal_x / tz, J02 = -focal_x * txc / (tz * tz);
    float J11 = focal_y / tz, J12 = -focal_y * tyc / (tz * tz);

    // Wc = viewmatrix[:3,:3].T  -> Wc[a][b] = vm[b*4+a]
    float Wc00 = vm[0], Wc01 = vm[4], Wc02 = vm[8];
    float Wc10 = vm[1], Wc11 = vm[5], Wc12 = vm[9];
    float Wc20 = vm[2], Wc21 = vm[6], Wc22 = vm[10];

    float T200 = J00*Wc00 + J02*Wc20;
    float T201 = J00*Wc01 + J02*Wc21;
    float T202 = J00*Wc02 + J02*Wc22;
    float T210 = J11*Wc10 + J12*Wc20;
    float T211 = J11*Wc11 + J12*Wc21;
    float T212 = J11*Wc12 + J12*Wc22;

    float A0 = T200*Sg00 + T201*Sg01 + T202*Sg02;
    float A1 = T200*Sg01 + T201*Sg11 + T202*Sg12;
    float A2 = T200*Sg02 + T201*Sg12 + T202*Sg22;
    float B0 = T210*Sg00 + T211*Sg01 + T212*Sg02;
    float B1 = T210*Sg01 + T211*Sg11 + T212*Sg12;
    float B2 = T210*Sg02 + T211*Sg12 + T212*Sg22;
    float cov00 = A0*T200 + A1*T201 + A2*T202;
    float cov01 = A0*T210 + A1*T211 + A2*T212;
    float cov11 = B0*T210 + B1*T211 + B2*T212;

    float aa = cov00 + 0.3f, cc = cov11 + 0.3f, bb = cov01;
    float det = aa*cc - bb*bb;
    float det_s = (det == 0.0f) ? 1.0f : det;
    float conA = cc / det_s, conB = -bb / det_s, conC = aa / det_s;
    float mid  = 0.5f * (aa + cc);
    float disc = sqrtf(fmaxf(0.1f, mid*mid - det));
    float radius = ceilf(3.0f * sqrtf(fmaxf(mid + disc, mid - disc)));

    float gx = ((pp0 + 1.0f) * width  - 1.0f) * 0.5f;
    float gy = ((pp1 + 1.0f) * height - 1.0f) * 0.5f;

    bool vis = (depth > 0.2f) && (det != 0.0f) && (radius > 0.0f);
    out[NPIX*4 + i]      = vis ? radius : 0.0f;   // radii
    out[NPIX*4 + PN + i] = vis ? 1.0f   : 0.0f;   // visibility

    // SH evaluation (deg from input; deg == 3 here)
    float dxn = mx - campos[0], dyn = my - campos[1], dzn = mz - campos[2];
    float dn = sqrtf(dxn*dxn + dyn*dyn + dzn*dzn);
    dxn /= dn; dyn /= dn; dzn /= dn;
    int deg = deg_p[0];
    const float* sh = shs + (size_t)i * 48;
    const float C2_0 = 1.0925484305920792f,  C2_1 = -1.0925484305920792f;
    const float C2_2 = 0.31539156525252005f, C2_3 = -1.0925484305920792f, C2_4 = 0.5462742152960396f;
    const float C3_0 = -0.5900435899266435f, C3_1 = 2.890611442640554f,  C3_2 = -0.4570457994644658f;
    const float C3_3 = 0.3731763325901154f,  C3_4 = -0.4570457994644658f, C3_5 = 1.445305721320277f;
    const float C3_6 = -0.5900435899266435f;
    float res[3];
    #pragma unroll
    for (int ch = 0; ch < 3; ++ch) {
        float r = C0f * sh[0 + ch];
        if (deg >= 1)
            r += -C1f*dyn*sh[3+ch] + C1f*dzn*sh[6+ch] - C1f*dxn*sh[9+ch];
        if (deg >= 2) {
            float xx = dxn*dxn, yy = dyn*dyn, zz = dzn*dzn;
            float xy = dxn*dyn, yz = dyn*dzn, xz = dxn*dzn;
            r += C2_0*xy*sh[12+ch] + C2_1*yz*sh[15+ch] + C2_2*(2.0f*zz - xx - yy)*sh[18+ch]
               + C2_3*xz*sh[21+ch] + C2_4*(xx - yy)*sh[24+ch];
            if (deg >= 3) {
                r += C3_0*dyn*(3.0f*xx - yy)*sh[27+ch]
                   + C3_1*xy*dzn*sh[30+ch]
                   + C3_2*dyn*(4.0f*zz - xx - yy)*sh[33+ch]
                   + C3_3*dzn*(2.0f*zz - 3.0f*xx - 3.0f*yy)*sh[36+ch]
                   + C3_4*dxn*(4.0f*zz - xx - yy)*sh[39+ch]
                   + C3_5*dzn*(xx - yy)*sh[42+ch]
                   + C3_6*dxn*(xx - 3.0f*yy)*sh[45+ch];
            }
        }
        res[ch] = fmaxf(r + 0.5f, 0.0f);
    }

    // power(pix) = f0*px^2 + f1*py^2 + f2*px*py + f3*px + f4*py + f5
    float* featU  = ws;
    float* blendU = ws + (size_t)PN*8;
    float* dkey   = ws + (size_t)PN*32;
    featU[i*8+0] = -0.5f * conA;
    featU[i*8+1] = -0.5f * conC;
    featU[i*8+2] = -conB;
    featU[i*8+3] = conA*gx + conB*gy;
    featU[i*8+4] = conC*gy + conB*gx;
    featU[i*8+5] = -0.5f*(conA*gx*gx + conC*gy*gy) - conB*gx*gy;
    featU[i*8+6] = 0.0f;
    featU[i*8+7] = 0.0f;
    blendU[i*8+0] = opac[i];
    blendU[i*8+1] = res[0];
    blendU[i*8+2] = res[1];
    blendU[i*8+3] = res[2];
    blendU[i*8+4] = depth;
    blendU[i*8+5] = vis ? 1.0f : 0.0f;
    blendU[i*8+6] = 0.0f;
    blendU[i*8+7] = 0.0f;
    dkey[i] = vis ? depth : __builtin_inff();
}

// ---------------------------------------------------------------------------
// Single-workgroup bitonic sort of (depth,index) pairs (stable-equivalent via
// index tiebreak, matching jnp.argsort), then gather feat/blend into sorted
// order.  2048 keys in LDS (16 KB of the 320 KB WGP pool).
// ---------------------------------------------------------------------------
__global__ void __launch_bounds__(1024) sort_gather_kernel(float* __restrict__ ws)
{
    __shared__ float skey[PN];
    __shared__ int   sidx[PN];
    const float* dkey = ws + (size_t)PN*32;
    int tid = threadIdx.x;
    for (int i = tid; i < PN; i += 1024) { skey[i] = dkey[i]; sidx[i] = i; }
    __syncthreads();

    for (int k = 2; k <= PN; k <<= 1) {
        for (int j = k >> 1; j > 0; j >>= 1) {
            for (int i = tid; i < PN; i += 1024) {
                int ixj = i ^ j;
                if (ixj > i) {
                    bool asc = ((i & k) == 0);
                    float k1 = skey[i], k2 = skey[ixj];
                    int   i1 = sidx[i], i2 = sidx[ixj];
                    bool gt = (k1 > k2) || ((k1 == k2) && (i1 > i2));
                    if (gt == asc) {
                        skey[i] = k2; skey[ixj] = k1;
                        sidx[i] = i2; sidx[ixj] = i1;
                    }
                }
            }
            __syncthreads();
        }
    }

    float* featU  = ws;
    float* blendU = ws + (size_t)PN*8;
    float* featS  = ws + (size_t)PN*16;
    float* blendS = ws + (size_t)PN*24;
    for (int s = tid; s < PN; s += 1024) {
        int g = sidx[s];
        #pragma unroll
        for (int q = 0; q < 8; ++q) {
            featS[s*8+q]  = featU[g*8+q];
            blendS[s*8+q] = blendU[g*8+q];
        }
    }
}

// ---------------------------------------------------------------------------
// Render: each wave owns 32 pixels (two 16-pixel WMMA tiles).  Per 16-gaussian
// chunk: power tile via v_wmma_f32_16x16x4_f32 (K=6 padded to 8 -> 2 chained
// WMMAs per tile), ds_bpermute xor-16 exchange so each lane owns one pixel's
// 16 powers in sorted order, then exact front-to-back blend recurrence.
// ---------------------------------------------------------------------------
__device__ __forceinline__ float xchg16(float v)
{
    int src = (((int)(threadIdx.x & 31u)) ^ 16) << 2;  // byte addr of partner lane
    return __int_as_float(__builtin_amdgcn_ds_bpermute(src, __float_as_int(v)));
}

__global__ void __launch_bounds__(256) render_kernel(
    const float* __restrict__ featS, const float* __restrict__ blendS,
    const float* __restrict__ bg, float* __restrict__ out)
{
    int lane = (int)(threadIdx.x & 31u);
    int wave = (int)(threadIdx.x >> 5);
    int base = (blockIdx.x * 8 + wave) * 32;   // 32 pixels per wave
    int col  = lane & 15;
    int half = lane >> 4;

    // pixel monomial features [px^2, py^2, px*py, px, py, 1, 0, 0]
    float fA[8] = {0,0,0,0,0,0,0,0};
    float fB[8] = {0,0,0,0,0,0,0,0};
    {
        int pA = base + col;
        float px = (float)(pA / HIMG), py = (float)(pA % HIMG);
        fA[0]=px*px; fA[1]=py*py; fA[2]=px*py; fA[3]=px; fA[4]=py; fA[5]=1.0f;
        int pB = base + 16 + col;
        px = (float)(pB / HIMG); py = (float)(pB % HIMG);
        fB[0]=px*px; fB[1]=py*py; fB[2]=px*py; fB[3]=px; fB[4]=py; fB[5]=1.0f;
    }
    // B operand (4x16, N=pixel): VGPR0/1 hold K pairs split across lane halves
    v2f bA0 = { fA[2*half],     fA[2*half + 1] };
    v2f bA1 = { fA[4 + 2*half], fA[4 + 2*half + 1] };
    v2f bB0 = { fB[2*half],     fB[2*half + 1] };
    v2f bB1 = { fB[4 + 2*half], fB[4 + 2*half + 1] };

    float T = 1.0f, Tfin = 1.0f;
    float cr = 0.0f, cg = 0.0f, cb = 0.0f, cd = 0.0f;

    for (int cbase = 0; cbase < PN; cbase += 16) {
        // A operand (16x4, M=gaussian): lane holds gaussian cbase+col
        const float* fg = featS + (size_t)(cbase + col) * 8;
        v2f a0 = { fg[2*half],     fg[2*half + 1] };
        v2f a1 = { fg[4 + 2*half], fg[4 + 2*half + 1] };

        v8f dA = {}; v8f dB = {};
        dA = __builtin_amdgcn_wmma_f32_16x16x4_f32(false, a0, false, bA0, (short)0, dA, false, false);
        dA = __builtin_amdgcn_wmma_f32_16x16x4_f32(false, a1, false, bA1, (short)0, dA, false, false);
        dB = __builtin_amdgcn_wmma_f32_16x16x4_f32(false, a0, false, bB0, (short)0, dB, false, false);
        dB = __builtin_amdgcn_wmma_f32_16x16x4_f32(false, a1, false, bB1, (short)0, dB, false, false);

        // D layout: lane L -> pixel N=L%16; VGPR j -> gaussian j + 8*(L/16).
        // Exchange across lane^16 so lane L ends up owning pixel base+L with
        // powers for all 16 gaussians of this chunk, in sorted order.
        float pw[16];
        #pragma unroll
        for (int j = 0; j < 8; ++j) {
            float oA = xchg16(dA[j]);
            float oB = xchg16(dB[j]);
            pw[j]     = (half == 0) ? dA[j] : oB;
            pw[j + 8] = (half == 0) ? oA    : dB[j];
        }

        // sequential front-to-back blend (exact reference semantics)
        #pragma unroll
        for (int n = 0; n < 16; ++n) {
            const float* bd = blendS + (size_t)(cbase + n) * 8;  // wave-uniform
            float opa = bd[0], r = bd[1], g = bd[2], b = bd[3], dp = bd[4], vf = bd[5];
            float p = pw[n];
            float alpha = fminf(0.99f, opa * expf(p));
            bool valid = (vf != 0.0f) && (p <= 0.0f) && (alpha >= (1.0f / 255.0f));
            float ae = valid ? alpha : 0.0f;
            float Tn = T * (1.0f - ae);
            if (Tn >= 1.0e-4f) {
                float w = ae * T;        // alpha * T_excl
                cr += w * r; cg += w * g; cb += w * b; cd += w * dp;
                Tfin = Tn;               // last kept T_incl == min of kept
            }
            T = Tn;
        }
        // wave-uniform early exit: T monotonically non-increasing
        if (!__any(T >= 1.0e-4f)) break;
    }

    int pix = base + lane;  // lane owns pixel base+lane after exchange
    out[pix*3 + 0]    = cr + Tfin * bg[0];
    out[pix*3 + 1]    = cg + Tfin * bg[1];
    out[pix*3 + 2]    = cb + Tfin * bg[2];
    out[NPIX*3 + pix] = cd;
}

// ---------------------------------------------------------------------------
extern "C" void kernel_launch(void* const* d_in, const int* in_sizes, int n_in,
                              void* d_out, int out_size, void* d_ws, size_t ws_size,
                              hipStream_t stream)
{
    const int*   degp    = (const int*)  d_in[1];
    const float* bg      = (const float*)d_in[3];
    const int*   wp      = (const int*)  d_in[4];
    const int*   hp      = (const int*)  d_in[5];
    const float* means3D = (const float*)d_in[6];
    const float* shs     = (const float*)d_in[7];
    const float* opac    = (const float*)d_in[8];
    const float* scales  = (const float*)d_in[9];
    const float* smod    = (const float*)d_in[10];
    const float* rots    = (const float*)d_in[11];
    const float* vm      = (const float*)d_in[12];
    const float* pm      = (const float*)d_in[13];
    const float* campos  = (const float*)d_in[14];
    const float* tfx     = (const float*)d_in[15];
    const float* tfy     = (const float*)d_in[16];

    float* ws  = (float*)d_ws;
    float* out = (float*)d_out;

    preprocess_kernel<<<PN / 256, 256, 0, stream>>>(
        means3D, shs, opac, scales, smod, rots, vm, pm, campos,
        tfx, tfy, degp, wp, hp, ws, out);

    sort_gather_kernel<<<1, 1024, 0, stream>>>(ws);

    render_kernel<<<NPIX / 32 / 8, 256, 0, stream>>>(
        ws + (size_t)PN * 16, ws + (size_t)PN * 24, bg, out);
}